// LSTM_DIST_19292993093819
// MI455X (gfx1250) — compile-verified
//
#include <hip/hip_runtime.h>

// LSTM-minus span representation for MI455X (gfx1250).
// Pure gather + subtract: ~128 MB gathered reads + 96 MB write-once output
// => ~10us at 23.3 TB/s HBM. No contraction => WMMA inapplicable; the win
// is memory-path shape:
//  - 1 block (128 thr = 4 wave32) per span: span indices are block-uniform
//    -> s_load_b128, and all index math / boundary masks stay scalar.
//  - 32-bit scalar index clamping (s_min/s_max) instead of 64-bit VALU.
//  - Unconditional clamped gathers + branchless zero-select for boundary
//    cases (uniform s_cselect, no exec-mask churn).
//  - float4 lanes -> global_load_b128 / global_store_b128, fully coalesced.
//  - Non-temporal stores for the write-once output so it doesn't evict
//    reusable lstm_out rows from the 192 MB L2.

typedef __attribute__((ext_vector_type(4))) float v4f;

namespace {
constexpr int kB  = 32;
constexpr int kT  = 4096;
constexpr int kH2 = 1024;   // 2H
constexpr int kH  = 512;    // H
constexpr int kS  = 512;
constexpr int kVec  = 4;
constexpr int kRowV = kH2 / kVec;  // 256 float4 per [b,t,:] row
constexpr int kHV   = kH  / kVec;  // 128 float4 per half-row
}

__global__ __launch_bounds__(128) void minus_span_kernel(
    const float* __restrict__ lstm_out,      // [B, T, 2H] f32
    const long long* __restrict__ spans,     // [B, S, 2] i64
    float* __restrict__ out)                 // [B, S, 3H] f32
{
    const int bs = blockIdx.x;               // span id in [0, B*S)
    const int c  = threadIdx.x;              // float4 chunk in [0, kHV)
    const int b  = bs >> 9;                  // bs / S (S = 512)

    // Block-uniform -> scalar s_load; truncate to i32 immediately (T = 4096).
    const int i = (int)spans[2 * (size_t)bs + 0];
    const int j = (int)spans[2 * (size_t)bs + 1];

    // 32-bit scalar clamps (s_max_i32 / s_min_i32).
    const int jc  = min(max(j, 0), kT - 1);          // fwd[j]
    const int ic  = min(max(i, 0), kT - 1);          // bwd[i]
    const int ipc = min(max(i - 1, 0), kT - 1);      // fwd[i-1] (clamped)
    const int jpc = min(max(j + 1, 0), kT - 1);      // bwd[j+1] (clamped)

    const bool has_pre  = (i - 1) >= 0;
    const bool has_post = (j + 1) < kT;
    const bool skip     = (i == 0) && (j == 0);

    const v4f* base = (const v4f*)lstm_out + (size_t)b * kT * kRowV;
    const v4f zero = {0.0f, 0.0f, 0.0f, 0.0f};

    // Unconditional clamped gathers: scalar base + (c*16B) lane offset.
    v4f f_end   = base[(size_t)jc  * kRowV +       c];   // fwd[j]
    v4f b_start = base[(size_t)ic  * kRowV + kHV + c];   // bwd[i]
    v4f f_pre   = base[(size_t)ipc * kRowV +       c];   // fwd[i-1]
    v4f b_post  = base[(size_t)jpc * kRowV + kHV + c];   // bwd[j+1]

    // Branchless boundary masking (uniform predicates -> scalar cselect).
    if (!has_pre)  f_pre  = zero;
    if (!has_post) b_post = zero;

    v4f o0 = f_end   - f_pre;    // [f_j - f_{i-1}]
    v4f o1 = b_start - b_post;   // [b_i - b_{j+1}]
    v4f o2 = f_pre   - b_post;   // [f_{i-1} - b_{j+1}]

    if (skip) { o0 = zero; o1 = zero; o2 = zero; }

    // Write-once output: 128-bit non-temporal stores (TH=NT).
    v4f* obase = (v4f*)out + (size_t)bs * (3 * kHV);
    __builtin_nontemporal_store(o0, obase + c);
    __builtin_nontemporal_store(o1, obase + kHV + c);
    __builtin_nontemporal_store(o2, obase + 2 * kHV + c);
}

extern "C" void kernel_launch(void* const* d_in, const int* in_sizes, int n_in,
                              void* d_out, int out_size, void* d_ws, size_t ws_size,
                              hipStream_t stream) {
    const float*     lstm_out = (const float*)d_in[0];      // [32,4096,1024] f32
    const long long* spans    = (const long long*)d_in[1];  // [32,512,2] i64
    float*           out      = (float*)d_out;              // [32,512,1536] f32

    (void)in_sizes; (void)n_in; (void)out_size; (void)d_ws; (void)ws_size;

    const int num_spans = kB * kS;   // 16384 blocks, 1 span each
    minus_span_kernel<<<num_spans, 128, 0, stream>>>(lstm_out, spans, out);
}